// GNNat_14190571946497
// MI455X (gfx1250) — compile-verified
//
#include <hip/hip_runtime.h>
#include <hip/hip_bf16.h>

#define N_NODES   50000
#define N_EDGES   1600000
#define POS_DIM   64
#define IN_FEATS  128
#define OUT_FEATS 128
#define ROW_STR   (POS_DIM + IN_FEATS)   // 192 floats per row of x and out
#define MLP_DIM   32

typedef __attribute__((ext_vector_type(2))) float v2f;
typedef __attribute__((ext_vector_type(8))) float v8f;

// ---------------------------------------------------------------------------
// init: zero m_i, deg=1 (self loop), copy coors -> out[:, :64], fold edge MLP
// into two scalars A,B (pe = sigmoid(dist*A + B)).
// ---------------------------------------------------------------------------
__global__ void k_init(float* __restrict__ mi, float* __restrict__ deg,
                       float* __restrict__ scal,
                       const float* __restrict__ mw1, const float* __restrict__ mb1,
                       const float* __restrict__ mw2, const float* __restrict__ mb2,
                       const float* __restrict__ x, float* __restrict__ out) {
    size_t tid = (size_t)blockIdx.x * blockDim.x + threadIdx.x;
    size_t nth = (size_t)gridDim.x * blockDim.x;

    for (size_t i = tid; i < (size_t)N_NODES * OUT_FEATS; i += nth) mi[i] = 0.0f;
    for (size_t i = tid; i < N_NODES; i += nth) deg[i] = 1.0f;   // self-loop

    // coors copy: 16 float4 per row, row stride 48 float4 in both x and out
    const float4* xv = (const float4*)x;
    float4*       ov = (float4*)out;
    for (size_t i = tid; i < (size_t)N_NODES * 16; i += nth) {
        size_t node = i >> 4, d4 = i & 15;
        ov[node * 48 + d4] = xv[node * 48 + d4];
    }

    if (tid == 0) {
        float A = 0.0f, B = 0.0f;
        for (int k = 0; k < MLP_DIM; ++k) { A += mw1[k] * mw2[k]; B += mb1[k] * mw2[k]; }
        scal[0] = A;
        scal[1] = B + mb2[0];
    }
}

// ---------------------------------------------------------------------------
// in-degree by col (self-loop already baked into deg init)
// ---------------------------------------------------------------------------
__global__ void k_degree(const int* __restrict__ ei, float* __restrict__ deg) {
    long e = (long)blockIdx.x * blockDim.x + threadIdx.x;
    if (e < N_EDGES) atomicAdd(&deg[ei[N_EDGES + e]], 1.0f);
}

__global__ void k_dinv(const float* __restrict__ deg, float* __restrict__ dinv) {
    int i = blockIdx.x * blockDim.x + threadIdx.x;
    if (i < N_NODES) {
        float d = deg[i];
        dinv[i] = d > 0.0f ? rsqrtf(d) : 0.0f;
    }
}

// ---------------------------------------------------------------------------
// one wave32 per edge: 64-d squared distance (2 floats/lane + xor reduce),
// scalar gate, then float4 of feats per lane -> 4 f32 atomics into m_i[col].
// Self-loops are edges [N_EDGES, N_EDGES+N_NODES) with r==c (dist==0).
// ---------------------------------------------------------------------------
__global__ __launch_bounds__(256) void k_edge(const int* __restrict__ ei,
                                              const float* __restrict__ x,
                                              const float* __restrict__ dinv,
                                              const float* __restrict__ scal,
                                              float* __restrict__ mi) {
    const long ETOT = (long)N_EDGES + N_NODES;
    int  lane = threadIdx.x & 31;
    long e    = (long)blockIdx.x * 8 + (threadIdx.x >> 5);
    if (e >= ETOT) return;

    int r, c;
    if (e < N_EDGES) { r = ei[e]; c = ei[N_EDGES + e]; }
    else             { r = c = (int)(e - N_EDGES); }

    const float* xr = x + (size_t)r * ROW_STR;
    const float* xc = x + (size_t)c * ROW_STR;

    float2 pr = *(const float2*)(xr + 2 * lane);
    float2 pc = *(const float2*)(xc + 2 * lane);
    float dx = pr.x - pc.x, dy = pr.y - pc.y;
    float d  = dx * dx + dy * dy;
#pragma unroll
    for (int off = 16; off; off >>= 1) d += __shfl_xor(d, off, 32);

    float z  = d * scal[0] + scal[1];
    float pe = 1.0f / (1.0f + expf(-z));
    float w  = pe * dinv[r] * dinv[c];

    float4 f = *(const float4*)(xr + POS_DIM + 4 * lane);
    float* dst = mi + (size_t)c * OUT_FEATS + 4 * lane;
    atomicAdd(dst + 0, w * f.x);
    atomicAdd(dst + 1, w * f.y);
    atomicAdd(dst + 2, w * f.z);
    atomicAdd(dst + 3, w * f.w);
}

// ---------------------------------------------------------------------------
// hidden = m_i @ w_no + bias via V_WMMA_F32_16X16X4_F32.
// One wave per 16x16 output tile; 8 waves/block cover the 128 output cols of
// one 16-row stripe. 50000 rows = 3125 * 16 exactly.
// A 16x4 layout: lane%16 = M row; lane/16 selects K pair {0,1}/{2,3}.
// B 4x16 layout (symmetric): lane%16 = N col; lane/16 selects K pair.
// C/D: acc[i] = D[(lane/16)*8 + i][lane%16].
// ---------------------------------------------------------------------------
__global__ __launch_bounds__(256) void k_gemm(const float* __restrict__ mi,
                                              const float* __restrict__ w,
                                              const float* __restrict__ bias,
                                              float* __restrict__ out) {
    int wid   = threadIdx.x >> 5;     // 0..7 -> N tile
    int lane  = threadIdx.x & 31;
    int half  = lane >> 4;            // 0/1 -> K sub-pair, D row half
    int l16   = lane & 15;
    int mtile = blockIdx.x;           // 0..3124
    int n     = wid * 16 + l16;       // output column 0..127
    int m_a   = mtile * 16 + l16;     // A row for this lane

    float bv = bias[n];
    v8f acc;
#pragma unroll
    for (int i = 0; i < 8; ++i) acc[i] = bv;

    const float* arow = mi + (size_t)m_a * IN_FEATS + half * 2;          // a[k0] = m_i[m_a][k0 + 2*half]
    const float* bcol = w + (size_t)(half * 2) * OUT_FEATS + n;          // b at [k0 + 2*half][n]

#pragma unroll 4
    for (int k0 = 0; k0 < IN_FEATS; k0 += 4) {
        v2f a, b;
        a.x = arow[k0];
        a.y = arow[k0 + 1];
        b.x = bcol[(size_t)k0 * OUT_FEATS];
        b.y = bcol[(size_t)(k0 + 1) * OUT_FEATS];
        acc = __builtin_amdgcn_wmma_f32_16x16x4_f32(
            /*neg_a=*/false, a, /*neg_b=*/false, b,
            /*c_mod=*/(short)0, acc, /*reuse_a=*/false, /*reuse_b=*/false);
    }

    int mbase = mtile * 16 + half * 8;
#pragma unroll
    for (int i = 0; i < 8; ++i)
        out[(size_t)(mbase + i) * ROW_STR + POS_DIM + n] = acc[i];
}

// ---------------------------------------------------------------------------
extern "C" void kernel_launch(void* const* d_in, const int* in_sizes, int n_in,
                              void* d_out, int out_size, void* d_ws, size_t ws_size,
                              hipStream_t stream) {
    const float* x    = (const float*)d_in[0];
    const int*   ei   = (const int*)d_in[1];
    const float* w_no = (const float*)d_in[2];
    const float* bias = (const float*)d_in[3];
    const float* mw1  = (const float*)d_in[4];
    const float* mb1  = (const float*)d_in[5];
    const float* mw2  = (const float*)d_in[6];
    const float* mb2  = (const float*)d_in[7];
    float* out = (float*)d_out;

    float* ws   = (float*)d_ws;
    float* scal = ws;                                   // 2 floats (padded)
    float* deg  = ws + 64;                              // N floats
    float* dinv = deg + N_NODES;                        // N floats
    size_t mi_off = ((size_t)(64 + 2 * N_NODES) + 63) & ~(size_t)63;
    float* mi   = ws + mi_off;                          // N * 128 floats (~25.6 MB)

    k_init<<<25000, 256, 0, stream>>>(mi, deg, scal, mw1, mb1, mw2, mb2, x, out);
    k_degree<<<(N_EDGES + 255) / 256, 256, 0, stream>>>(ei, deg);
    k_dinv<<<(N_NODES + 255) / 256, 256, 0, stream>>>(deg, dinv);

    long etot   = (long)N_EDGES + N_NODES;              // incl. self loops
    int  eblks  = (int)((etot + 7) / 8);                // 8 waves/block, 1 edge/wave
    k_edge<<<eblks, 256, 0, stream>>>(ei, x, dinv, scal, mi);

    k_gemm<<<N_NODES / 16, 256, 0, stream>>>(mi, w_no, bias, out);
}